// SRFSmile_5145370821355
// MI455X (gfx1250) — compile-verified
//
#include <hip/hip_runtime.h>
#include <hip/hip_bf16.h>
#include <math.h>

// ---------------------------------------------------------------------------
// Problem constants (fixed by the reference setup_inputs):
//   x_blur            : (B=4, K=8, H=1024, W=1024) f32
//   smile_shift_col   : (4, 8, 64) f32
//   smile_temp_col    : (4, 8, 64) f32
//   srf_base_logits   : (8, 8) f32
//   out               : (4, 8, 1024, 1024) f32
// out[b,m,h,c] = sum_k x[b,k,h,c] * W[b,m,k,c]
// ---------------------------------------------------------------------------

typedef __attribute__((ext_vector_type(2))) float v2f;
typedef __attribute__((ext_vector_type(8))) float v8f;

#define BB 4
#define BM 8
#define BH 8
#define HH 1024
#define WW 1024
#define WCS 64

#define HT 16   // h rows per block (WMMA N)
#define CT 32   // columns per block (= waves per block)

// ---------------------------------------------------------------------------
// Kernel 1: materialize W[b][m][k][c]  (4*8*8*1024 floats = 1 MB in d_ws).
// One thread per (b,m,c).
// ---------------------------------------------------------------------------
__global__ void srf_build_weights(const float* __restrict__ shift_col,
                                  const float* __restrict__ temp_col,
                                  const float* __restrict__ srf_logits,
                                  float* __restrict__ wout) {
    int idx = blockIdx.x * blockDim.x + threadIdx.x;   // (b*BM + m)*WW + c
    if (idx >= BB * BM * WW) return;
    int c  = idx & (WW - 1);
    int bm = idx >> 10;           // b*BM + m
    int m  = bm & (BM - 1);

    // srf = softmax(srf_base_logits[m, :])  (8 values, L2-resident)
    float l[BH];
    float mx = -1e30f;
    #pragma unroll
    for (int j = 0; j < BH; ++j) { l[j] = srf_logits[m * BH + j]; mx = fmaxf(mx, l[j]); }
    float s = 0.f;
    #pragma unroll
    for (int j = 0; j < BH; ++j) { l[j] = expf(l[j] - mx); s += l[j]; }
    float inv = 1.f / s;
    #pragma unroll
    for (int j = 0; j < BH; ++j) l[j] *= inv;

    // linear upsample (align_corners=False) of shift/temp at column c
    float src = (c + 0.5f) * ((float)WCS / (float)WW) - 0.5f;
    src = fmaxf(src, 0.0f);
    float x0f = floorf(src);
    int   x0  = min((int)x0f, WCS - 1);
    int   x1  = min(x0 + 1, WCS - 1);
    float fr  = src - x0f;
    const float* sp = shift_col + bm * WCS;
    const float* tp = temp_col + bm * WCS;
    float shv = sp[x0] * (1.f - fr) + sp[x1] * fr;
    float tv  = tp[x0] * (1.f - fr) + tp[x1] * fr;

    // grid-sample in y (border clamp, align_corners=True) + log/temp
    float lg[BH];
    float lmx = -1e30f;
    #pragma unroll
    for (int k = 0; k < BH; ++k) {
        float gy  = (-1.f + 2.f * (float)k / (float)(BH - 1)) + shv * (2.f / (float)(BH - 1));
        float y   = fminf(fmaxf((gy + 1.f) * 0.5f * (float)(BH - 1), 0.f), (float)(BH - 1));
        float y0f = floorf(y);
        int   y0  = (int)y0f;
        int   y1  = min(y0 + 1, BH - 1);
        float f2  = y - y0f;
        float sh  = l[y0] * (1.f - f2) + l[y1] * f2;
        lg[k] = logf(sh + 1e-6f) / (tv + 1e-6f);
        lmx = fmaxf(lmx, lg[k]);
    }
    float ss = 0.f;
    #pragma unroll
    for (int k = 0; k < BH; ++k) { lg[k] = expf(lg[k] - lmx); ss += lg[k]; }
    float sinv = 1.f / ss;
    #pragma unroll
    for (int k = 0; k < BH; ++k)
        wout[((bm * BH + k) << 10) + c] = lg[k] * sinv;
}

// ---------------------------------------------------------------------------
// Kernel 2: streaming einsum via V_WMMA_F32_16X16X4_F32.
// Block = 1024 threads (32 waves). Block tile: (b, 16 h, 32 c).
// Wave w owns column c0+w: D(16m x 16h) = W(16m x 8k) * X(8k x 16h),
// done as two chained 16x16x4 f32 WMMAs. Global traffic is 128-bit
// coalesced both ways; the per-column transpose happens in LDS.
// ---------------------------------------------------------------------------
__global__ void __launch_bounds__(1024)
srf_mix_wmma(const float* __restrict__ x,
             const float* __restrict__ wts,
             float* __restrict__ out) {
    __shared__ float s_x[BH][HT][36];   // [k][h][c]  pad 36 -> conflict-light ds reads
    __shared__ float s_w[BM][BH][33];   // [m][k][c]
    __shared__ float s_o[BM][HT][36];   // [m][h][c]

    const int tid = threadIdx.x;
    const int b  = blockIdx.z;
    const int h0 = blockIdx.y * HT;
    const int c0 = blockIdx.x * CT;

    // ---- stage weights tile (2048 floats, coalesced 128B rows) ----
    #pragma unroll
    for (int r = 0; r < 2; ++r) {
        int e = tid + r * 1024;
        int c = e & 31, k = (e >> 5) & 7, m = e >> 8;
        s_w[m][k][c] = wts[(((b * BM + m) * BH + k) << 10) + c0 + c];
    }
    // ---- stage x tile: 1024 x float4 = 4096 floats, global_load_b128 ----
    {
        int c4 = tid & 7, h = (tid >> 3) & 15, k = tid >> 7;
        const float4 v = *reinterpret_cast<const float4*>(
            x + ((((b * BH + k) << 10) + h0 + h) << 10) + c0 + (c4 << 2));
        *reinterpret_cast<float4*>(&s_x[k][h][c4 << 2]) = v;
    }
    __syncthreads();

    // ---- per-wave WMMA: one column each ----
    const int wave = tid >> 5;       // column within tile
    const int lane = tid & 31;
    const int lm   = lane & 15;      // A: row m  /  B: col h
    const int hi   = lane >> 4;      // selects K=2,3 half per ISA 16x4 layout

    v8f acc = {};
    #pragma unroll
    for (int kb = 0; kb < BH; kb += 4) {
        const int ka = kb + (hi << 1);
        v2f a, bv;
        a.x  = (lm < BM) ? s_w[lm][ka][wave]     : 0.0f;   // A[m][ka]
        a.y  = (lm < BM) ? s_w[lm][ka + 1][wave] : 0.0f;   // A[m][ka+1]
        bv.x = s_x[ka][lm][wave];                          // B[ka][h]
        bv.y = s_x[ka + 1][lm][wave];                      // B[ka+1][h]
        acc = __builtin_amdgcn_wmma_f32_16x16x4_f32(
            /*neg_a=*/false, a, /*neg_b=*/false, bv,
            /*c_mod=*/(short)0, acc, /*reuse_a=*/false, /*reuse_b=*/false);
    }

    // ---- D -> LDS transpose: VGPR r, lanes 0-15 hold (M=r, N=lane) ----
    if (lane < 16) {
        #pragma unroll
        for (int r = 0; r < BM; ++r)
            s_o[r][lm][wave] = acc[r];
    }
    __syncthreads();

    // ---- coalesced 128-bit store ----
    {
        int c4 = tid & 7, h = (tid >> 3) & 15, m = tid >> 7;
        float4 v = *reinterpret_cast<const float4*>(&s_o[m][h][c4 << 2]);
        *reinterpret_cast<float4*>(
            out + ((((b * BM + m) << 10) + h0 + h) << 10) + c0 + (c4 << 2)) = v;
    }
}

// ---------------------------------------------------------------------------
extern "C" void kernel_launch(void* const* d_in, const int* in_sizes, int n_in,
                              void* d_out, int out_size, void* d_ws, size_t ws_size,
                              hipStream_t stream) {
    const float* x_blur    = (const float*)d_in[0];
    const float* shift_col = (const float*)d_in[1];
    const float* temp_col  = (const float*)d_in[2];
    const float* srf_log   = (const float*)d_in[3];
    float* wts = (float*)d_ws;                 // 4*8*8*1024 floats = 1 MB
    float* out = (float*)d_out;

    srf_build_weights<<<(BB * BM * WW + 255) / 256, 256, 0, stream>>>(
        shift_col, temp_col, srf_log, wts);

    dim3 grid(WW / CT, HH / HT, BB);           // (32, 64, 4)
    srf_mix_wmma<<<grid, 1024, 0, stream>>>(x_blur, wts, out);
}